// GCNMixEncoder_1443109012139
// MI455X (gfx1250) — compile-verified
//
#include <hip/hip_runtime.h>
#include <stdint.h>

// ---------------- problem constants (match reference) ----------------
#define USER_COUNT 100000
#define ITEM_COUNT 50000
#define N_NODES    (USER_COUNT + ITEM_COUNT)
#define EMB        128
#define N_LAYERS   3
#define BATCH      4096

// Native clang vector (works with __builtin_nontemporal_*, unlike HIP float4)
typedef float v4f __attribute__((ext_vector_type(4)));

// ---------------- CDNA5 helpers ----------------

// Async global -> LDS copy of one dword per active lane (gfx1250 ASYNCcnt path).
// VDST = VGPR holding LDS byte address, VADDR = 64-bit global address.
__device__ __forceinline__ void async_g2l_b32(uint32_t lds_addr, const void* gaddr) {
    asm volatile("global_load_async_to_lds_b32 %0, %1, off"
                 :: "v"(lds_addr), "v"((uint64_t)(uintptr_t)gaddr)
                 : "memory");
}

__device__ __forceinline__ void wait_async0() {
#if defined(__has_builtin) && __has_builtin(__builtin_amdgcn_s_wait_asynccnt)
    __builtin_amdgcn_s_wait_asynccnt(0);
#else
    asm volatile("s_wait_asynccnt 0" ::: "memory");
#endif
    asm volatile("" ::: "memory");
}

// Hardware f32 scatter-add, no-return form (RMW resolves at L2, no VGPR
// writeback, no CAS loop). VGLOBAL opcode 86: global_atomic_add_f32.
__device__ __forceinline__ void atomic_add_f32(float* p, float v) {
    asm volatile("global_atomic_add_f32 %0, %1, off"
                 :: "v"((uint64_t)(uintptr_t)p), "v"(v)
                 : "memory");
}

// ---------------- kernels (k_spmm first so the disasm snippet shows it) ----

// Edge-parallel SpMM: y[rows[e]] += vals[e] * x[cols[e]]  (rows of 128 f32).
// One wave32 owns exactly one chunk of 32 edges (grid sized to cover nnz):
//   - rows/cols/vals for the chunk are staged into LDS with async-to-LDS
//   - each lane prefetches all 4 cachelines of its own edge's source row
//   - for each edge j: wave cooperatively loads the 512B source row as
//     v4f/lane and issues 4 global_atomic_add_f32 per lane into y.
__launch_bounds__(256)
__global__ void k_spmm(const float* __restrict__ vals, const int* __restrict__ rows,
                       const int* __restrict__ cols, const float* __restrict__ x,
                       float* __restrict__ y, int nnz) {
    __shared__ int sh[8 * 96];                 // per-wave: 32 rows | 32 cols | 32 val bits
    const int lane = threadIdx.x & 31;
    const int wib  = threadIdx.x >> 5;
    int* wsh = &sh[wib * 96];

    const int base = (blockIdx.x * 8 + wib) * 32;   // chunk start edge
    if (base >= nnz) return;
    const int jmax = min(32, nnz - base);

    if (lane < jmax) {
        const int e = base + lane;
        async_g2l_b32((uint32_t)(uintptr_t)&wsh[lane],      rows + e);
        async_g2l_b32((uint32_t)(uintptr_t)&wsh[32 + lane], cols + e);
        async_g2l_b32((uint32_t)(uintptr_t)&wsh[64 + lane], vals + e);
    }
    wait_async0();

    // Prefetch the entire 512B source row of this lane's edge (4 lines).
    if (lane < jmax) {
        const float* rowp = x + (size_t)wsh[32 + lane] * EMB;
        __builtin_prefetch(rowp +  0, 0, 0);
        __builtin_prefetch(rowp + 32, 0, 0);
        __builtin_prefetch(rowp + 64, 0, 0);
        __builtin_prefetch(rowp + 96, 0, 0);
    }

    for (int j = 0; j < jmax; ++j) {
        const int   r = wsh[j];
        const int   c = wsh[32 + j];
        const float v = __int_as_float(wsh[64 + j]);
        const v4f xv = *(const v4f*)(x + (size_t)c * EMB + lane * 4);
        float* dst = y + (size_t)r * EMB + lane * 4;
        atomic_add_f32(dst + 0, v * xv.x);
        atomic_add_f32(dst + 1, v * xv.y);
        atomic_add_f32(dst + 2, v * xv.z);
        atomic_add_f32(dst + 3, v * xv.w);
    }
}

// acc = cur = concat(user_emb, item_emb), in 16B strides.
// acc is streamed (non-temporal) so it does not pollute L2; cur is reused
// immediately by the first SpMM so it stays temporal (L2-resident).
__global__ void k_init(const float* __restrict__ ue, const float* __restrict__ ie,
                       float* __restrict__ acc, float* __restrict__ cur) {
    size_t i = (size_t)blockIdx.x * blockDim.x + threadIdx.x;
    const size_t n4 = (size_t)N_NODES * (EMB / 4);
    if (i >= n4) return;
    const size_t u4 = (size_t)USER_COUNT * (EMB / 4);
    v4f v = (i < u4) ? ((const v4f*)ue)[i] : ((const v4f*)ie)[i - u4];
    __builtin_nontemporal_store(v, (v4f*)acc + i);
    ((v4f*)cur)[i] = v;
}

// next = 0 with regular-temporal stores: pins the scatter destination in L2
// before the atomic storm.
__global__ void k_zero(float* __restrict__ p) {
    size_t i = (size_t)blockIdx.x * blockDim.x + threadIdx.x;
    const size_t n4 = (size_t)N_NODES * (EMB / 4);
    if (i >= n4) return;
    v4f z = {0.f, 0.f, 0.f, 0.f};
    ((v4f*)p)[i] = z;
}

// acc += x. acc traffic is streaming-only -> fully non-temporal so the
// L2 keeps holding cur/next. x (== next) is L2-hot from the atomics.
__global__ void k_accadd(float* __restrict__ acc, const float* __restrict__ x) {
    size_t i = (size_t)blockIdx.x * blockDim.x + threadIdx.x;
    const size_t n4 = (size_t)N_NODES * (EMB / 4);
    if (i >= n4) return;
    v4f a = __builtin_nontemporal_load((const v4f*)acc + i);
    v4f b = ((const v4f*)x)[i];
    a = a + b;
    __builtin_nontemporal_store(a, (v4f*)acc + i);
}

// out[0:4096]    = acc[users]            * 0.25
// out[4096:8192] = acc[USER_COUNT+items] * 0.25
__global__ void k_gather(const float* __restrict__ acc, const int* __restrict__ users,
                         const int* __restrict__ items, float* __restrict__ out) {
    const int b = blockIdx.x;          // 0 .. 2*BATCH-1
    const int t = threadIdx.x;         // 0 .. 127
    long long row;
    if (b < BATCH) row = (long long)users[b];
    else           row = (long long)USER_COUNT + items[b - BATCH];
    out[(size_t)b * EMB + t] = acc[row * (long long)EMB + t] * 0.25f;
}

// ---------------- launcher ----------------

extern "C" void kernel_launch(void* const* d_in, const int* in_sizes, int n_in,
                              void* d_out, int out_size, void* d_ws, size_t ws_size,
                              hipStream_t stream) {
    const float* ue    = (const float*)d_in[0];   // [100000,128]
    const float* ie    = (const float*)d_in[1];   // [50000,128]
    const float* vals  = (const float*)d_in[2];   // [nnz]
    const int*   rows  = (const int*)d_in[3];     // [nnz]
    const int*   cols  = (const int*)d_in[4];     // [nnz]
    const int*   users = (const int*)d_in[5];     // [4096]
    const int*   items = (const int*)d_in[6];     // [4096]
    float* out = (float*)d_out;
    const int nnz = in_sizes[2];

    const size_t nodeElems = (size_t)N_NODES * EMB;     // 19.2M floats
    float* acc  = (float*)d_ws;                         // 76.8 MB
    float* bufA = acc + nodeElems;                      // 76.8 MB
    float* bufB = bufA + nodeElems;                     // 76.8 MB (ws >= 230.4 MB)

    const int thr  = 256;
    const size_t n4 = nodeElems / 4;
    const int blk4 = (int)((n4 + thr - 1) / thr);       // 18750

    k_init<<<blk4, thr, 0, stream>>>(ue, ie, acc, bufA);

    float* cur = bufA;
    float* nxt = bufB;
    const int wavesPerBlock = 256 / 32;
    const int chunks = (nnz + 31) / 32;                 // 50000
    const int spmmBlocks = (chunks + wavesPerBlock - 1) / wavesPerBlock; // 6250

    for (int l = 0; l < N_LAYERS; ++l) {
        k_zero<<<blk4, thr, 0, stream>>>(nxt);
        k_spmm<<<spmmBlocks, 256, 0, stream>>>(vals, rows, cols, cur, nxt, nnz);
        k_accadd<<<blk4, thr, 0, stream>>>(acc, nxt);
        float* t = cur; cur = nxt; nxt = t;
    }

    k_gather<<<2 * BATCH, EMB, 0, stream>>>(acc, users, items, out);
}